// VectorQuantize2_19696720019634
// MI455X (gfx1250) — compile-verified
//
#include <hip/hip_runtime.h>

typedef __attribute__((ext_vector_type(2))) float v2f;
typedef __attribute__((ext_vector_type(8))) float v8f;

#define N_ROWS     8192      // B*H*W
#define CDIM       256       // channels = K
#define NE         16384     // codebook size
#define HW         1024      // H*W
#define SPLIT      2         // split-K over the code dimension
#define ROWS_PER_WG 128      // 8 waves * 16 rows
#define TILE_J     16        // codes per WMMA tile
#define LDS_STRIDE 260       // padded row stride (floats) -> conflict-free banks

// workspace layout (float offsets)
#define WS_CN    0                       // 16384 : code norms
#define WS_PMIN  (WS_CN   + NE)          // 16384 : split-K partial min values
#define WS_PIDX  (WS_PMIN + N_ROWS*SPLIT)// 16384 : split-K partial argmin (int)
#define WS_IDXF  (WS_PIDX + N_ROWS*SPLIT)//  8192 : final indices (int)
#define WS_PART  (WS_IDXF + N_ROWS)      //  8192 : per-block loss partials
// total = 65536 floats = 256 KB

#define OUT_XQ   0
#define OUT_LOSS 2097152
#define OUT_CODE 2097153

// ---------------- kernel 1: codebook row norms (wave per code) ----------------
__global__ __launch_bounds__(256) void vq_norms(const float* __restrict__ w,
                                                float* __restrict__ cn) {
  const int lane = threadIdx.x & 31, wid = threadIdx.x >> 5;
  const int code = blockIdx.x * 8 + wid;
  const float* r = w + (size_t)code * CDIM;
  float s = 0.f;
#pragma unroll
  for (int q = 0; q < 8; ++q) { float v = r[q * 32 + lane]; s = fmaf(v, v, s); }
#pragma unroll
  for (int m = 16; m >= 1; m >>= 1) s += __shfl_xor(s, m, 32);
  if (lane == 0) cn[code] = s;
}

// ---------------- kernel 2: WMMA distance GEMM + running argmin ----------------
__global__ __launch_bounds__(256) void vq_main(const float* __restrict__ x,
                                               const float* __restrict__ w,
                                               const float* __restrict__ cn,
                                               float* __restrict__ pmin,
                                               int* __restrict__ pidx) {
  __shared__ float sB[TILE_J * LDS_STRIDE];

  const int t     = threadIdx.x;
  const int lane  = t & 31, wid = t >> 5;
  const int mrow  = lane & 15;          // M (and B-column) for this lane
  const int khalf = lane >> 4;          // selects K pair {0,1} vs {2,3}
  const int rowbase = blockIdx.x * ROWS_PER_WG;
  const int s       = blockIdx.y;       // split-K half of the codebook
  const int row     = rowbase + wid * 16 + mrow;

  // ---- preload A fragment: this wave's 16 rows, full K=256, WMMA A layout ----
  // flat[n][c] = x[(n>>10)*C*HW + c*HW + (n&1023)]
  const float* xb = x + (size_t)(row >> 10) * CDIM * HW + (row & (HW - 1));
  float a0[64], a1[64];
#pragma unroll
  for (int kk = 0; kk < 64; ++kk) {
    const int c0 = kk * 4 + 2 * khalf;
    a0[kk] = xb[(size_t)c0 * HW];
    a1[kk] = xb[(size_t)(c0 + 1) * HW];
  }

  float rmin[8]; int ridx[8];
#pragma unroll
  for (int i = 0; i < 8; ++i) { rmin[i] = 3.4e38f; ridx[i] = 0; }

  // cooperative B-tile staging: 256 threads x 16 floats = 16x256 tile
  const int jr      = t >> 4;           // code row within tile
  const int colbase = (t & 15) * 16;    // 16 consecutive K per thread
  const int jbase   = s * (NE / SPLIT);
  const int ntiles  = (NE / SPLIT) / TILE_J;   // 512

  float4 bt[4];
  {
    const float* src = w + (size_t)(jbase + jr) * CDIM + colbase;
#pragma unroll
    for (int q = 0; q < 4; ++q) bt[q] = ((const float4*)src)[q];
  }

  for (int tile = 0; tile < ntiles; ++tile) {
    __syncthreads();
#pragma unroll
    for (int q = 0; q < 4; ++q)
      *(float4*)&sB[jr * LDS_STRIDE + colbase + q * 4] = bt[q];
    __syncthreads();

    if (tile + 1 < ntiles) {  // prefetch next tile into registers (overlaps WMMA)
      const float* src = w + (size_t)(jbase + (tile + 1) * TILE_J + jr) * CDIM + colbase;
#pragma unroll
      for (int q = 0; q < 4; ++q) bt[q] = ((const float4*)src)[q];
    }

    const int j0 = jbase + tile * TILE_J;

    v8f acc = {};
#pragma unroll
    for (int kk = 0; kk < 64; ++kk) {
      v2f av; av.x = a0[kk]; av.y = a1[kk];
      const float* bs = &sB[mrow * LDS_STRIDE + kk * 4 + 2 * khalf];
      v2f bv; bv.x = bs[0]; bv.y = bs[1];
      acc = __builtin_amdgcn_wmma_f32_16x16x4_f32(
          /*neg_a=*/false, av, /*neg_b=*/false, bv,
          /*c_mod=*/(short)0, acc, /*reuse_a=*/false, /*reuse_b=*/false);
    }

    // score = ||w_j||^2 - 2 * x.w_j   (||x||^2 is row-constant -> drops from argmin)
    const int   jcol = j0 + mrow;
    const float cnv  = cn[jcol];
#pragma unroll
    for (int i = 0; i < 8; ++i) {
      const float sc = fmaf(-2.f, acc[i], cnv);
      if (sc < rmin[i]) { rmin[i] = sc; ridx[i] = jcol; }
    }
  }

  // cross-lane argmin reduce over the 16 columns held in each half-wave
#pragma unroll
  for (int i = 0; i < 8; ++i) {
    float v = rmin[i]; int id = ridx[i];
#pragma unroll
    for (int m = 8; m >= 1; m >>= 1) {
      const float ov = __shfl_xor(v, m, 32);
      const int   oi = __shfl_xor(id, m, 32);
      if (ov < v || (ov == v && oi < id)) { v = ov; id = oi; }
    }
    if (mrow == 0) {   // lanes 0 (M=i) and 16 (M=i+8) hold the result
      const int r = rowbase + wid * 16 + i + 8 * khalf;
      pmin[r * SPLIT + s] = v;
      pidx[r * SPLIT + s] = id;
    }
  }
}

// ---------------- kernel 3: merge split-K partials, emit x_code ----------------
__global__ __launch_bounds__(256) void vq_combine(const float* __restrict__ pmin,
                                                  const int* __restrict__ pidx,
                                                  int* __restrict__ idxf,
                                                  float* __restrict__ out) {
  const int n = blockIdx.x * blockDim.x + threadIdx.x;
  const float m0 = pmin[n * 2 + 0], m1 = pmin[n * 2 + 1];
  const int   i0 = pidx[n * 2 + 0], i1 = pidx[n * 2 + 1];
  const int id = (m1 < m0) ? i1 : i0;   // tie -> lower half -> lower index (argmin semantics)
  idxf[n] = id;
  out[OUT_CODE + n] = (float)id;
}

// ---------------- kernel 4: gather xq, write xq_img, loss partials ----------------
__global__ __launch_bounds__(256) void vq_gather(const float* __restrict__ x,
                                                 const float* __restrict__ w,
                                                 const int* __restrict__ idxf,
                                                 float* __restrict__ out,
                                                 float* __restrict__ part) {
  const int tx = threadIdx.x;            // 32 hw lanes (coalesced dim)
  const int ty = threadIdx.y;            // 8 channels
  const int hw = blockIdx.x * 32 + tx;
  const int c  = blockIdx.y * 8 + ty;
  const int b  = blockIdx.z;
  const int n  = b * HW + hw;

  const int id = idxf[n];
  const size_t xo = ((size_t)b * CDIM + c) * HW + hw;
  const float xv = x[xo];
  const float wv = w[(size_t)id * CDIM + c];
  out[OUT_XQ + xo] = xv + (wv - xv);     // straight-through value, same rounding as ref
  const float d = wv - xv;
  float ssum = d * d;
#pragma unroll
  for (int m = 16; m >= 1; m >>= 1) ssum += __shfl_xor(ssum, m, 32);

  __shared__ float sw[8];
  if (tx == 0) sw[ty] = ssum;
  __syncthreads();
  if (tx == 0 && ty == 0) {
    float tot = 0.f;
#pragma unroll
    for (int q = 0; q < 8; ++q) tot += sw[q];
    part[blockIdx.x + 32 * (blockIdx.y + 32 * blockIdx.z)] = tot;
  }
}

// ---------------- kernel 5: deterministic final loss reduction ----------------
__global__ __launch_bounds__(256) void vq_loss(const float* __restrict__ part,
                                               float* __restrict__ out) {
  const int t = threadIdx.x;
  float s = 0.f;
  for (int k = 0; k < 32; ++k) s += part[t + k * 256];   // fixed order
  __shared__ float sm[256];
  sm[t] = s;
  __syncthreads();
  for (int off = 128; off >= 1; off >>= 1) {
    if (t < off) sm[t] += sm[t + off];
    __syncthreads();
  }
  if (t == 0) out[OUT_LOSS] = sm[0] * (1.25f / 2097152.f);
}

extern "C" void kernel_launch(void* const* d_in, const int* in_sizes, int n_in,
                              void* d_out, int out_size, void* d_ws, size_t ws_size,
                              hipStream_t stream) {
  const float* x = (const float*)d_in[0];   // [8,256,32,32] fp32
  const float* w = (const float*)d_in[1];   // [16384,256]   fp32
  float* out = (float*)d_out;
  float* ws  = (float*)d_ws;

  float* cn   = ws + WS_CN;
  float* pmin = ws + WS_PMIN;
  int*   pidx = (int*)(ws + WS_PIDX);
  int*   idxf = (int*)(ws + WS_IDXF);
  float* part = ws + WS_PART;

  vq_norms<<<NE / 8, 256, 0, stream>>>(w, cn);

  dim3 gmain(N_ROWS / ROWS_PER_WG, SPLIT);          // 64 x 2 WGs = 1024 waves
  vq_main<<<gmain, 256, 0, stream>>>(x, w, cn, pmin, pidx);

  vq_combine<<<N_ROWS / 256, 256, 0, stream>>>(pmin, pidx, idxf, out);

  dim3 gg(HW / 32, CDIM / 8, 8), bb(32, 8);
  vq_gather<<<gg, bb, 0, stream>>>(x, w, idxf, out, part);

  vq_loss<<<1, 256, 0, stream>>>(part, out);
}